// Set2Set_9122510537145
// MI455X (gfx1250) — compile-verified
//
#include <hip/hip_runtime.h>
#include <math.h>

// ---------------------------------------------------------------------------
// Set2Set (MPNN global pooling) for MI455X / gfx1250.
//   N = 1,000,000 nodes, D = 128, B = 16384 graphs, 6 iterations.
// Bandwidth-bound on streaming feat (512MB/iter) -> one-pass online softmax.
// LSTM GEMM ([B,384]@[384,512] fp32) done with V_WMMA_F32_16X16X4_F32.
// ---------------------------------------------------------------------------

typedef __attribute__((ext_vector_type(2))) float v2f;
typedef __attribute__((ext_vector_type(8))) float v8f;

#define DH    128            // hidden dim (specialized; WMMA tiling assumes 128)
#define KDIM  384            // 2*DH (q_star) + DH (h)
#define GDIM  512            // 4*DH gate columns
#define NITER 6

// ---------------------------------------------------------------- utilities
__device__ __forceinline__ float sigm_f(float x) {
    return 1.0f / (1.0f + __expf(-x));
}
__device__ __forceinline__ float tanh_f(float x) {
    x = fminf(15.0f, fmaxf(-15.0f, x));
    float e = __expf(2.0f * x);
    return (e - 1.0f) / (e + 1.0f);
}

__global__ void zero_kernel(float* __restrict__ p, int n) {
    int i = blockIdx.x * blockDim.x + threadIdx.x;
    if (i < n) p[i] = 0.0f;
}

// seg_start[g] = lower_bound(segment_ids, g); segment_ids is sorted.
__global__ void seg_start_kernel(const int* __restrict__ seg, int n, int b1,
                                 int* __restrict__ out) {
    int g = blockIdx.x * blockDim.x + threadIdx.x;
    if (g >= b1) return;
    int lo = 0, hi = n;
    while (lo < hi) {
        int mid = (lo + hi) >> 1;
        if (seg[mid] < g) lo = mid + 1; else hi = mid;
    }
    out[g] = lo;
}

// Pack W = [W_ih.T ; W_hh.T]  (KDIM x GDIM) into pair-interleaved layout:
//   Wtp[(kk>>1)*2*GDIM + col*2 + (kk&1)]
// so a lane's B-fragment (two consecutive K for one column) is one b64 load.
// Also fold the two bias vectors.
__global__ void pack_weights_kernel(const float* __restrict__ W_ih,   // [512,256]
                                    const float* __restrict__ W_hh,   // [512,128]
                                    const float* __restrict__ b_ih,
                                    const float* __restrict__ b_hh,
                                    float* __restrict__ Wtp,
                                    float* __restrict__ biasv) {
    int idx = blockIdx.x * blockDim.x + threadIdx.x;
    if (idx < KDIM * GDIM) {
        int kk  = idx / GDIM;
        int col = idx % GDIM;
        float w = (kk < 2 * DH) ? W_ih[(size_t)col * (2 * DH) + kk]
                                : W_hh[(size_t)col * DH + (kk - 2 * DH)];
        Wtp[(size_t)(kk >> 1) * (2 * GDIM) + col * 2 + (kk & 1)] = w;
    }
    if (idx < GDIM) biasv[idx] = b_ih[idx] + b_hh[idx];
}

// ---------------------------------------------------------------- LSTM step
// gates = [q_star, h_in] @ W + bias ; i,f,g,o activations ; update c, h_out.
// Grid: (B/16) blocks x 256 threads (8 waves). Wave w owns gate-column group
// 16*w..16*w+15 and accumulates all four gate tiles for the same 16x16 slice,
// so the nonlinearity + cell update is fully in-register.
// A fragment layout (ISA 32-bit A 16x4): lanes 0-15 rows M=0..15 K=k,k+1 ;
// lanes 16-31 rows M=0..15 K=k+2,k+3. C layout: VGPR v -> M=v (+8 hi lanes),
// N = lane&15.
__global__ __launch_bounds__(256)
void lstm_kernel(const float* __restrict__ q_star,  // [B,256]
                 const float* __restrict__ h_in,    // [B,128]
                 float*       __restrict__ c,       // [B,128] (in-place: unique owner per elem)
                 const float* __restrict__ Wtp,     // [192][512][2]
                 const float* __restrict__ biasv,   // [512]
                 float*       __restrict__ h_out) { // [B,128]
    const int wave  = threadIdx.x >> 5;     // 0..7 : column group
    const int lane  = threadIdx.x & 31;
    const int l15   = lane & 15;
    const int khalf = (lane < 16) ? 0 : 2;
    const int m     = blockIdx.x * 16;

    const int rowA = m + l15;
    const float* qrow = q_star + (size_t)rowA * (2 * DH);
    const float* hrow = h_in   + (size_t)rowA * DH;

    const int col0 = 0 * DH + wave * 16 + l15;   // i
    const int col1 = 1 * DH + wave * 16 + l15;   // f
    const int col2 = 2 * DH + wave * 16 + l15;   // g
    const int col3 = 3 * DH + wave * 16 + l15;   // o

    v8f acc0 = {}, acc1 = {}, acc2 = {}, acc3 = {};

    for (int k = 0; k < KDIM; k += 4) {
        const int kb = k + khalf;                       // even; wave-uniform branch
        v2f a;
        if (kb < 2 * DH) a = *(const v2f*)(qrow + kb);
        else             a = *(const v2f*)(hrow + (kb - 2 * DH));
        const float* wbase = Wtp + (size_t)(kb >> 1) * (2 * GDIM);
        v2f b0 = *(const v2f*)(wbase + col0 * 2);
        v2f b1 = *(const v2f*)(wbase + col1 * 2);
        v2f b2 = *(const v2f*)(wbase + col2 * 2);
        v2f b3 = *(const v2f*)(wbase + col3 * 2);
        acc0 = __builtin_amdgcn_wmma_f32_16x16x4_f32(false, a, false, b0, (short)0, acc0, false, false);
        acc1 = __builtin_amdgcn_wmma_f32_16x16x4_f32(false, a, false, b1, (short)0, acc1, false, false);
        acc2 = __builtin_amdgcn_wmma_f32_16x16x4_f32(false, a, false, b2, (short)0, acc2, false, false);
        acc3 = __builtin_amdgcn_wmma_f32_16x16x4_f32(false, a, false, b3, (short)0, acc3, false, false);
    }

    const float bi = biasv[col0], bf = biasv[col1];
    const float bg = biasv[col2], bo = biasv[col3];
    const int colg  = wave * 16 + l15;
    const int rbase = m + ((lane < 16) ? 0 : 8);

#pragma unroll
    for (int v = 0; v < 8; ++v) {
        const int row = rbase + v;
        const size_t cidx = (size_t)row * DH + colg;
        const float gi = sigm_f(acc0[v] + bi);
        const float gf = sigm_f(acc1[v] + bf);
        const float gg = tanh_f(acc2[v] + bg);
        const float go = sigm_f(acc3[v] + bo);
        const float cn = gf * c[cidx] + gi * gg;
        c[cidx]     = cn;
        h_out[cidx] = go * tanh_f(cn);
    }
}

// ------------------------------------------------------------ attention pass
// One 128-thread block (4 waves) per graph; nodes of a graph are contiguous.
// Single pass over feat with an online softmax: per-wave running (m, s, r),
// merged through LDS at the end. readout = r_tot / s_tot.
// Writes q_star = [h | readout] directly (this is d_out).
__global__ __launch_bounds__(128)
void attn_kernel(const float* __restrict__ feat,       // [N,128]
                 const int*   __restrict__ seg_start,  // [B+1]
                 const float* __restrict__ h,          // [B,128] == q
                 float*       __restrict__ q_star) {   // [B,256]
    __shared__ float m_s[4];
    __shared__ float s_s[4];
    __shared__ __align__(16) float r_s[4 * DH];

    const int g     = blockIdx.x;
    const int w     = threadIdx.x >> 5;
    const int lane  = threadIdx.x & 31;
    const int dbase = 4 * lane;

    const int start = seg_start[g];
    const int end   = seg_start[g + 1];

    const float4 qv = *(const float4*)(h + (size_t)g * DH + dbase);

    float  m_run = -INFINITY;
    float  s_run = 0.0f;
    float4 r     = make_float4(0.f, 0.f, 0.f, 0.f);

    for (int i = start + w; i < end; i += 4) {
        if (i + 4 < end)
            __builtin_prefetch(feat + (size_t)(i + 4) * DH + dbase, 0, 0);
        const float4 f = *(const float4*)(feat + (size_t)i * DH + dbase);
        float p = f.x * qv.x + f.y * qv.y + f.z * qv.z + f.w * qv.w;
        p += __shfl_xor(p, 1,  32);
        p += __shfl_xor(p, 2,  32);
        p += __shfl_xor(p, 4,  32);
        p += __shfl_xor(p, 8,  32);
        p += __shfl_xor(p, 16, 32);          // e replicated across the wave
        const float mn = fmaxf(m_run, p);
        const float sc = __expf(m_run - mn); // first node: exp(-inf)=0
        const float ex = __expf(p - mn);
        r.x = r.x * sc + ex * f.x;
        r.y = r.y * sc + ex * f.y;
        r.z = r.z * sc + ex * f.z;
        r.w = r.w * sc + ex * f.w;
        s_run = s_run * sc + ex;
        m_run = mn;
    }

    if (lane == 0) { m_s[w] = m_run; s_s[w] = s_run; }
    *(float4*)(r_s + w * DH + dbase) = r;
    __syncthreads();

    const int t = threadIdx.x;               // 0..127 : output dim
    float mstar = -INFINITY;
#pragma unroll
    for (int ww = 0; ww < 4; ++ww)
        if (s_s[ww] > 0.0f) mstar = fmaxf(mstar, m_s[ww]);
    float stot = 0.0f, accv = 0.0f;
#pragma unroll
    for (int ww = 0; ww < 4; ++ww) {
        if (s_s[ww] > 0.0f) {
            const float sc = __expf(m_s[ww] - mstar);
            stot += s_s[ww] * sc;
            accv += r_s[ww * DH + t] * sc;
        }
    }
    const float ro = (stot > 0.0f) ? (accv / stot) : 0.0f;
    const size_t qb = (size_t)g * (2 * DH);
    q_star[qb + DH + t] = ro;                      // readout half
    q_star[qb + t]      = h[(size_t)g * DH + t];   // q half
}

// ---------------------------------------------------------------- launcher
extern "C" void kernel_launch(void* const* d_in, const int* in_sizes, int n_in,
                              void* d_out, int out_size, void* d_ws, size_t ws_size,
                              hipStream_t stream) {
    const float* feat = (const float*)d_in[0];
    const int*   seg  = (const int*)  d_in[1];
    // d_in[2] = num_graphs (device scalar) — derived from out_size instead.
    const float* W_ih = (const float*)d_in[3];
    const float* W_hh = (const float*)d_in[4];
    const float* b_ih = (const float*)d_in[5];
    const float* b_hh = (const float*)d_in[6];
    float* q_star = (float*)d_out;          // [B, 2*DH] lives in d_out

    const int n = in_sizes[0] / DH;         // node count
    const int B = out_size / (2 * DH);      // graph count

    // workspace carve-out (256B aligned)
    char*  ws  = (char*)d_ws;
    size_t off = 0;
    auto carve = [&](size_t bytes) -> void* {
        void* p = ws + off;
        off = (off + bytes + 255) & ~(size_t)255;
        return p;
    };
    float* h0     = (float*)carve((size_t)B * DH * sizeof(float));
    float* h1     = (float*)carve((size_t)B * DH * sizeof(float));
    float* cbuf   = (float*)carve((size_t)B * DH * sizeof(float));
    float* Wtp    = (float*)carve((size_t)KDIM * GDIM * sizeof(float));
    float* biasv  = (float*)carve((size_t)GDIM * sizeof(float));
    int*   segst  = (int*)  carve((size_t)(B + 1) * sizeof(int));
    (void)ws_size;

    const int nh = B * DH;
    zero_kernel<<<(nh + 255) / 256, 256, 0, stream>>>(h0, nh);
    zero_kernel<<<(nh + 255) / 256, 256, 0, stream>>>(cbuf, nh);
    zero_kernel<<<(out_size + 255) / 256, 256, 0, stream>>>(q_star, out_size);
    pack_weights_kernel<<<(KDIM * GDIM + 255) / 256, 256, 0, stream>>>(
        W_ih, W_hh, b_ih, b_hh, Wtp, biasv);
    seg_start_kernel<<<(B + 1 + 255) / 256, 256, 0, stream>>>(seg, n, B + 1, segst);

    float* hin = h0;
    float* hout = h1;
    for (int it = 0; it < NITER; ++it) {
        lstm_kernel<<<B / 16, 256, 0, stream>>>(q_star, hin, cbuf, Wtp, biasv, hout);
        attn_kernel<<<B, 128, 0, stream>>>(feat, segst, hout, q_star);
        float* tmp = hin; hin = hout; hout = tmp;
    }
}